// CustomAttention_29712583754406
// MI455X (gfx1250) — compile-verified
//
#include <hip/hip_runtime.h>

// Problem constants (B, S, H) from the reference.
constexpr int kB = 4;
constexpr int kS = 512;
constexpr int kH = 256;
constexpr int kM = kB * kS;          // 2048 rows for the projection GEMMs

typedef __attribute__((ext_vector_type(2))) float v2f;
typedef __attribute__((ext_vector_type(8))) float v8f;

// ---------------------------------------------------------------------------
// tanh: gfx1250 has a hardware transcendental tanh. Guarded fallback via exp.
// ---------------------------------------------------------------------------
__device__ __forceinline__ float dev_tanh(float x) {
#if __has_builtin(__builtin_amdgcn_tanhf)
    return __builtin_amdgcn_tanhf(x);
#else
    float ax = fabsf(x);
    float e  = __expf(-2.0f * ax);          // in (0,1], no overflow
    float t  = (1.0f - e) / (1.0f + e);
    return copysignf(t, x);
#endif
}

// ---------------------------------------------------------------------------
// Kernel 1: projection GEMM  Y(M=2048,N=256) = X(2048,256) * W(256,256) [+bias]
// One wave32 per 16x16 output tile, V_WMMA_F32_16X16X4_F32 over K.
// A 16x4 frag: lane L -> A[L&15][2*(L>>4) + {0,1}]
// B 4x16 frag: lane L -> B[2*(L>>4) + {0,1}][L&15]
// C/D: vgpr j -> row j + 8*(lane>>4), col lane&15
// ---------------------------------------------------------------------------
__global__ void __launch_bounds__(256)
proj_wmma_kernel(const float* __restrict__ X, const float* __restrict__ W,
                 const float* __restrict__ bias, float* __restrict__ Y) {
    const int lane = threadIdx.x & 31;
    const int wave = threadIdx.x >> 5;
    const int tile = blockIdx.x * 8 + wave;          // 2048 tiles total
    const int ntn  = kH / 16;                        // 16 tiles along N
    const int tm   = tile / ntn;                     // 0..127
    const int tn   = tile % ntn;                     // 0..15

    const int m   = (lane & 15);
    const int kk  = (lane >> 4) * 2;                 // 0 or 2
    const int row = tm * 16 + m;                     // A row for this lane
    const int col = tn * 16 + m;                     // B col for this lane

    v8f c = {};
    #pragma unroll 4
    for (int k0 = 0; k0 < kH; k0 += 4) {
        v2f a, b;
        const float* ap = X + (size_t)row * kH + (k0 + kk);
        a.x = ap[0];
        a.y = ap[1];
        b.x = W[(size_t)(k0 + kk) * kH + col];
        b.y = W[(size_t)(k0 + kk + 1) * kH + col];
        c = __builtin_amdgcn_wmma_f32_16x16x4_f32(false, a, false, b,
                                                  (short)0, c, false, false);
    }

    const int rbase = tm * 16 + 8 * (lane >> 4);
    const float bv  = bias ? bias[col] : 0.0f;
    #pragma unroll
    for (int j = 0; j < 8; ++j)
        Y[(size_t)(rbase + j) * kH + col] = c[j] + bv;
}

// ---------------------------------------------------------------------------
// Kernel 2: pair grid.  scores[b,t,t'] = sigmoid( sum_h Wa[h]*tanh(HT+HTP) + ba )
// HT already contains the +b_h fold. 16x32 pair tile per 256-thread block.
// ---------------------------------------------------------------------------
constexpr int TPM = 16;      // t rows per block
constexpr int TPN = 32;      // t' cols per block
constexpr int HPAD = kH + 1; // LDS pad: kills 32-way bank conflict on sB

__global__ void __launch_bounds__(256)
score_kernel(const float* __restrict__ HT, const float* __restrict__ HTP,
             const float* __restrict__ Wa, const float* __restrict__ ba,
             float* __restrict__ SC) {
    __shared__ float sA[TPM][HPAD];
    __shared__ float sB[TPN][HPAD];
    __shared__ float sW[kH];

    const int b   = blockIdx.z;
    const int t0  = blockIdx.y * TPM;
    const int p0  = blockIdx.x * TPN;
    const int tid = threadIdx.x;

    for (int i = tid; i < TPM * kH; i += 256)
        sA[i / kH][i % kH] = HT[((size_t)b * kS + t0 + i / kH) * kH + (i % kH)];
    for (int i = tid; i < TPN * kH; i += 256)
        sB[i / kH][i % kH] = HTP[((size_t)b * kS + p0 + i / kH) * kH + (i % kH)];
    for (int i = tid; i < kH; i += 256)
        sW[i] = Wa[i];
    __syncthreads();

    const float bav = ba[0];
    #pragma unroll
    for (int p = tid; p < TPM * TPN; p += 256) {      // 2 pairs per thread
        const int tm = p / TPN;
        const int tn = p % TPN;
        float acc = 0.0f;
        #pragma unroll 8
        for (int hh = 0; hh < kH; ++hh) {
            float v = sA[tm][hh] + sB[tn][hh];
            acc = __builtin_fmaf(sW[hh], dev_tanh(v), acc);
        }
        float s = acc + bav;
        s = 1.0f / (1.0f + __expf(-s));               // sigmoid
        SC[((size_t)b * kS + (t0 + tm)) * kS + (p0 + tn)] = s;
    }
}

// ---------------------------------------------------------------------------
// Kernel 3: softmax over t'. Input values are sigmoid outputs in (0,1), so
// exp() is unconditionally safe -> skip the max pass. One block per row.
// ---------------------------------------------------------------------------
__global__ void __launch_bounds__(256)
softmax_kernel(const float* __restrict__ SC, float* __restrict__ AW) {
    const size_t row = blockIdx.x;                     // b*S + t, 2048 rows
    const float* src = SC + row * kS;
    float*       dst = AW + row * kS;
    const int tid = threadIdx.x;

    float e0 = __expf(src[tid]);
    float e1 = __expf(src[tid + 256]);
    float sum = e0 + e1;

    #pragma unroll
    for (int off = 16; off > 0; off >>= 1)
        sum += __shfl_xor(sum, off, 32);

    __shared__ float wsum[8];
    if ((tid & 31) == 0) wsum[tid >> 5] = sum;
    __syncthreads();
    if (tid < 32) {
        float s = (tid < 8) ? wsum[tid] : 0.0f;
        #pragma unroll
        for (int off = 4; off > 0; off >>= 1)
            s += __shfl_xor(s, off, 32);
        if (tid == 0) wsum[0] = s;
    }
    __syncthreads();

    const float inv = 1.0f / wsum[0];
    dst[tid]       = e0 * inv;
    dst[tid + 256] = e1 * inv;
}

// ---------------------------------------------------------------------------
// Kernel 4: output GEMM per batch: O(512,256) = A(512,512) * h(512,256)
// Same wave-per-16x16-tile WMMA f32 structure, K = 512.
// ---------------------------------------------------------------------------
__global__ void __launch_bounds__(256)
out_wmma_kernel(const float* __restrict__ AW, const float* __restrict__ Hm,
                float* __restrict__ O) {
    const int lane = threadIdx.x & 31;
    const int wave = threadIdx.x >> 5;
    const int gt   = blockIdx.x * 8 + wave;           // 2048 tiles total
    const int tilesPerBatch = (kS / 16) * (kH / 16);  // 512
    const int b  = gt / tilesPerBatch;
    const int t  = gt % tilesPerBatch;
    const int tm = t / (kH / 16);
    const int tn = t % (kH / 16);

    const float* A  = AW + (size_t)b * kS * kS;
    const float* Bm = Hm + (size_t)b * kS * kH;

    const int m    = (lane & 15);
    const int kk   = (lane >> 4) * 2;
    const int mrow = tm * 16 + m;
    const int ncol = tn * 16 + m;

    v8f c = {};
    #pragma unroll 4
    for (int k0 = 0; k0 < kS; k0 += 4) {
        v2f a, bb;
        const float* ap = A + (size_t)mrow * kS + (k0 + kk);
        a.x  = ap[0];
        a.y  = ap[1];
        bb.x = Bm[(size_t)(k0 + kk) * kH + ncol];
        bb.y = Bm[(size_t)(k0 + kk + 1) * kH + ncol];
        c = __builtin_amdgcn_wmma_f32_16x16x4_f32(false, a, false, bb,
                                                  (short)0, c, false, false);
    }

    const int rbase = tm * 16 + 8 * (lane >> 4);
    float* Ob = O + (size_t)b * kS * kH;
    #pragma unroll
    for (int j = 0; j < 8; ++j)
        Ob[(size_t)(rbase + j) * kH + ncol] = c[j];
}

// ---------------------------------------------------------------------------
// Launch: d_out = [ output (B,S,H) | attention_weights (B,S,S) ]
// d_ws   = [ HT 2MB | HTP 2MB | scores 4MB ]
// ---------------------------------------------------------------------------
extern "C" void kernel_launch(void* const* d_in, const int* in_sizes, int n_in,
                              void* d_out, int out_size, void* d_ws, size_t ws_size,
                              hipStream_t stream) {
    const float* h   = (const float*)d_in[0];   // (B,S,H)
    const float* Wt  = (const float*)d_in[1];   // (H,H)
    const float* Wtp = (const float*)d_in[2];   // (H,H)
    const float* bh  = (const float*)d_in[3];   // (H)
    const float* Wa  = (const float*)d_in[4];   // (H)
    const float* ba  = (const float*)d_in[5];   // scalar

    float* out = (float*)d_out;                 // output first
    float* AW  = out + (size_t)kB * kS * kH;    // then attention weights

    float* HT  = (float*)d_ws;                  // (B*S, H), includes +b_h
    float* HTP = HT  + (size_t)kM * kH;         // (B*S, H)
    float* SC  = HTP + (size_t)kM * kH;         // (B, S, S) sigmoid scores

    // 1) projections (2048 WMMA tiles each -> 256 blocks of 8 waves)
    proj_wmma_kernel<<<256, 256, 0, stream>>>(h, Wt,  bh,      HT);
    proj_wmma_kernel<<<256, 256, 0, stream>>>(h, Wtp, nullptr, HTP);

    // 2) pair grid: sigmoid(Wa . tanh(HT + HTP) + ba)
    dim3 sg(kS / TPN, kS / TPM, kB);            // (16, 32, 4) = 2048 blocks
    score_kernel<<<sg, 256, 0, stream>>>(HT, HTP, Wa, ba, SC);

    // 3) softmax over t' (max pass skipped: inputs in (0,1))
    softmax_kernel<<<kB * kS, 256, 0, stream>>>(SC, AW);

    // 4) output = attention_weights @ h  (per-batch WMMA GEMM)
    out_wmma_kernel<<<256, 256, 0, stream>>>(AW, h, out);
}